// SSIMLoss_55731495633496
// MI455X (gfx1250) — compile-verified
//
#include <hip/hip_runtime.h>

// ---------------------------------------------------------------------------
// SSIM (window=11, sigma=1.5) on (32,3,512,512) f32 via separable Gaussian
// conv mapped onto V_WMMA_F32_16X16X4_F32 (CDNA5 / gfx1250, wave32).
// ---------------------------------------------------------------------------

typedef float v2f __attribute__((ext_vector_type(2)));
typedef float v8f __attribute__((ext_vector_type(8)));

#define WPB     8        // waves per block
#define IMG_H   512
#define IMG_W   512
#define TILE    16
#define RS      26       // region size (TILE + 10 halo)
#define XPITCH  28       // LDS pitch for image region (even -> 8B aligned v2f)
#define HPITCH  30       // LDS pitch for transposed H buffer (even, low-conflict)

// gaussian(11, 1.5), normalized (matches reference f64->f32 window)
__constant__ float GAUSS[11] = {
    0.001028380f, 0.007598758f, 0.036000773f, 0.109360619f, 0.213005542f,
    0.266011677f,
    0.213005542f, 0.109360619f, 0.036000773f, 0.007598758f, 0.001028380f
};

static __device__ __forceinline__ v8f wmma4(v2f a, v2f b, v8f c) {
    // D = A(16x4) * B(4x16) + C ; full f32
    return __builtin_amdgcn_wmma_f32_16x16x4_f32(
        /*neg_a=*/false, a, /*neg_b=*/false, b,
        /*c_mod=*/(short)0, c, /*reuse_a=*/false, /*reuse_b=*/false);
}

__global__ __launch_bounds__(256)
void ssim_tile_kernel(const float* __restrict__ img1,
                      const float* __restrict__ img2,
                      float* __restrict__ partial)
{
    __shared__ __align__(16) float sX[WPB][RS * XPITCH];   // 23.3 KB
    __shared__ __align__(16) float sY[WPB][RS * XPITCH];   // 23.3 KB
    __shared__ __align__(16) float sH[WPB][TILE * HPITCH]; // 15.4 KB (transposed H)
    __shared__ float sG[16];
    __shared__ float sPart[WPB];

    const int tid  = threadIdx.x;
    const int wave = tid >> 5;
    const int lane = tid & 31;

    if (tid < 16) sG[tid] = (tid < 11) ? GAUSS[tid] : 0.0f;
    __syncthreads();

    // tile assignment: 96 planes * 32*32 tiles, one tile per wave
    const int tileId = blockIdx.x * WPB + wave;
    const int plane  = tileId >> 10;           // / (32*32)
    const int rem    = tileId & 1023;
    const int r0     = (rem >> 5) << 4;        // tile row origin
    const int c0     = (rem & 31) << 4;        // tile col origin

    const float* __restrict__ p1 = img1 + (size_t)plane * (IMG_H * IMG_W);
    const float* __restrict__ p2 = img2 + (size_t)plane * (IMG_H * IMG_W);

    float* wx = sX[wave];
    float* wy = sY[wave];
    float* wh = sH[wave];

    // ---- stage 26x26 region (zero padded at image borders) ----
    for (int idx = lane; idx < RS * RS; idx += 32) {
        const int rr = idx / RS;
        const int cc = idx - rr * RS;
        const int gr = r0 - 5 + rr;
        const int gc = c0 - 5 + cc;
        float a = 0.0f, b = 0.0f;
        if ((unsigned)gr < (unsigned)IMG_H && (unsigned)gc < (unsigned)IMG_W) {
            const size_t o = (size_t)gr * IMG_W + gc;
            a = p1[o];
            b = p2[o];
        }
        wx[rr * XPITCH + cc] = a;
        wy[rr * XPITCH + cc] = b;
    }
    // zero pad cols 26,27 (read by K-chunk 6; weights there are 0, but avoid NaN*0)
    for (int i = lane; i < RS * 2; i += 32) {
        const int rr = i >> 1;
        const int cc = RS + (i & 1);
        wx[rr * XPITCH + cc] = 0.0f;
        wy[rr * XPITCH + cc] = 0.0f;
    }
    // zero H rows 26,27 (transposed layout: col n at n*HPITCH + row)
    {
        const int n  = lane >> 1;
        const int rr = RS + (lane & 1);
        wh[n * HPITCH + rr] = 0.0f;
    }

    const int m  = lane & 15;   // M (A rows / weight rows) and N (B cols) index
    const int kh = lane >> 4;   // lane-half selects K pair

    // ---- banded gaussian weights, per lane; shared by both passes ----
    // W[k, m] = g[k - m], k = 4t + 2*kh + j  (A layout and B layout both give
    // K = vgpr + 2*(lane>>4), M/N = lane&15)
    v2f wt[7];
#pragma unroll
    for (int t = 0; t < 7; ++t) {
#pragma unroll
        for (int j = 0; j < 2; ++j) {
            const int d = 4 * t + 2 * kh + j - m;
            wt[t][j] = (d >= 0 && d <= 10) ? sG[d] : 0.0f;
        }
    }

    // ---- horizontal pass: H = X * Wh ; 2 row-blocks (rows 0..15 / 10..25) ----
    v8f hx0 = {}, hx1 = {}, hy0 = {}, hy1 = {};
    v8f hxx0 = {}, hxx1 = {}, hyy0 = {}, hyy1 = {}, hxy0 = {}, hxy1 = {};
#pragma unroll
    for (int t = 0; t < 7; ++t) {
        const int cb = 4 * t + 2 * kh;
        const v2f ax0 = *(const v2f*)&wx[(0  + m) * XPITCH + cb];
        const v2f ax1 = *(const v2f*)&wx[(10 + m) * XPITCH + cb];
        const v2f ay0 = *(const v2f*)&wy[(0  + m) * XPITCH + cb];
        const v2f ay1 = *(const v2f*)&wy[(10 + m) * XPITCH + cb];
        const v2f b   = wt[t];
        hx0  = wmma4(ax0,        b, hx0);
        hx1  = wmma4(ax1,        b, hx1);
        hy0  = wmma4(ay0,        b, hy0);
        hy1  = wmma4(ay1,        b, hy1);
        hxx0 = wmma4(ax0 * ax0,  b, hxx0);
        hxx1 = wmma4(ax1 * ax1,  b, hxx1);
        hyy0 = wmma4(ay0 * ay0,  b, hyy0);
        hyy1 = wmma4(ay1 * ay1,  b, hyy1);
        hxy0 = wmma4(ax0 * ay0,  b, hxy0);
        hxy1 = wmma4(ax1 * ay1,  b, hxy1);
    }

    // ---- vertical pass: O = Wv * H ; single reused transposed H buffer ----
    auto vertical = [&](v8f a0, v8f a1) -> v8f {
        // C/D layout: VGPR v, lane -> row = v + 8*kh, col = m
#pragma unroll
        for (int v = 0; v < 8; ++v) {
            wh[m * HPITCH + (v + 8 * kh)]      = a0[v];   // H rows 0..15
            wh[m * HPITCH + (v + 8 * kh + 10)] = a1[v];   // H rows 10..25
        }
        v8f acc = {};
#pragma unroll
        for (int t = 0; t < 7; ++t) {
            const v2f bq = *(const v2f*)&wh[m * HPITCH + 4 * t + 2 * kh];
            acc = wmma4(wt[t], bq, acc);
        }
        return acc;
    };

    const v8f mu1 = vertical(hx0,  hx1);
    const v8f mu2 = vertical(hy0,  hy1);
    const v8f ex2 = vertical(hxx0, hxx1);
    const v8f ey2 = vertical(hyy0, hyy1);
    const v8f exy = vertical(hxy0, hxy1);

    // ---- SSIM map + reduction ----
    const float C1 = 0.0001f;   // 0.01^2
    const float C2 = 0.0009f;   // 0.03^2
    float local = 0.0f;
#pragma unroll
    for (int v = 0; v < 8; ++v) {
        const float m1  = mu1[v], m2 = mu2[v];
        const float m1s = m1 * m1;
        const float m2s = m2 * m2;
        const float m12 = m1 * m2;
        const float s1  = ex2[v] - m1s;
        const float s2  = ey2[v] - m2s;
        const float s12 = exy[v] - m12;
        const float num = (2.0f * m12 + C1) * (2.0f * s12 + C2);
        const float den = (m1s + m2s + C1) * (s1 + s2 + C2);
        local += num / den;
    }
#pragma unroll
    for (int off = 16; off > 0; off >>= 1)
        local += __shfl_xor(local, off, 32);
    if (lane == 0) sPart[wave] = local;
    __syncthreads();
    if (tid == 0) {
        float s = 0.0f;
#pragma unroll
        for (int w = 0; w < WPB; ++w) s += sPart[w];
        partial[blockIdx.x] = s;
    }
}

__global__ __launch_bounds__(256)
void ssim_reduce_kernel(const float* __restrict__ partial, int n,
                        float* __restrict__ out, float scale)
{
    __shared__ float sm[256];
    const int tid = threadIdx.x;
    float s = 0.0f;
    for (int i = tid; i < n; i += 256) s += partial[i];   // fixed-order strided sum
    sm[tid] = s;
    __syncthreads();
    for (int off = 128; off > 0; off >>= 1) {
        if (tid < off) sm[tid] += sm[tid + off];
        __syncthreads();
    }
    if (tid == 0) out[0] = sm[0] * scale;
}

extern "C" void kernel_launch(void* const* d_in, const int* in_sizes, int n_in,
                              void* d_out, int out_size, void* d_ws, size_t ws_size,
                              hipStream_t stream) {
    const float* img1 = (const float*)d_in[0];
    const float* img2 = (const float*)d_in[1];
    float* out        = (float*)d_out;
    float* partial    = (float*)d_ws;

    const long long total  = (long long)in_sizes[0];          // 32*3*512*512
    const int planes       = (int)(total / (IMG_H * IMG_W));  // 96
    const int nTiles       = planes * (IMG_H / TILE) * (IMG_W / TILE); // 98304
    const int blocks       = nTiles / WPB;                    // 12288

    ssim_tile_kernel<<<blocks, 256, 0, stream>>>(img1, img2, partial);
    ssim_reduce_kernel<<<1, 256, 0, stream>>>(partial, blocks, out,
                                              1.0f / (float)total);
}